// LUT_82085414961764
// MI455X (gfx1250) — compile-verified
//
#include <hip/hip_runtime.h>
#include <stdint.h>

// Piecewise-linear LUT eval: out = a[idx]*d + b[idx], idx = #(x < d), x sorted, K=1024.
// HBM-bound (512 MB stream @ 23.3 TB/s => ~22us floor). Strategy: bucketized search
// to cut LDS ops from ~13/elem to ~3/elem; LUT staged to LDS via gfx1250 async
// global->LDS copies; main stream uses non-temporal b128 load/store.

#define K_LUT 1024
#define NBINS 4096

typedef float v4f __attribute__((ext_vector_type(4)));
typedef int   v4i __attribute__((ext_vector_type(4)));

// Bucket function MUST be bit-identical between both kernels (it is: same inlined
// expression, no fma-contraction opportunity in (v-x0)*inv_w). Monotone in v, so
// bins[] brackets are exact regardless of rounding.
__device__ __forceinline__ int bucket_of(float v, float x0, float inv_w) {
  int t = (int)floorf((v - x0) * inv_w);
  t = t < 0 ? 0 : t;
  t = t > (NBINS - 1) ? (NBINS - 1) : t;
  return t;
}

// bins[j] = count of breakpoints i with bucket_of(x[i]) < j,  j in [0, NBINS].
// Guarded branchless binary search (11 steps); ~4K threads total, all L2-hit.
__global__ __launch_bounds__(256) void build_bins_kernel(const float* __restrict__ x,
                                                         int* __restrict__ bins) {
  int j = blockIdx.x * blockDim.x + threadIdx.x;
  if (j > NBINS) return;
  float x0 = x[0];
  float inv_w = (float)NBINS / (x[K_LUT - 1] - x0);
  int c = 0;
#pragma unroll
  for (int step = K_LUT; step >= 1; step >>= 1) {
    int nc = c + step;
    if (nc <= K_LUT) {
      float v = x[nc - 1];
      if (bucket_of(v, x0, inv_w) < j) c = nc;
    }
  }
  bins[j] = c;
}

// gfx1250 async global->LDS copies (ASYNCcnt-tracked). Per-lane LDS dest address
// in VDST vgpr, per-lane 64-bit global address in VADDR pair (GV mode).
__device__ __forceinline__ void async_ld_b128(uint32_t lds, const void* g) {
  asm volatile("global_load_async_to_lds_b128 %0, %1, off" ::"v"(lds), "v"(g) : "memory");
}
__device__ __forceinline__ void async_ld_b32(uint32_t lds, const void* g) {
  asm volatile("global_load_async_to_lds_b32 %0, %1, off" ::"v"(lds), "v"(g) : "memory");
}
// Low 32 bits of a flat shared address are the LDS byte offset (ISA 10.2).
__device__ __forceinline__ uint32_t lds_off(const void* p) {
  return (uint32_t)(uintptr_t)p;
}

__global__ __launch_bounds__(256) void lut_eval_kernel(
    const v4f* __restrict__ data4, v4f* __restrict__ out4, int nvec4,
    const float* __restrict__ x, const float* __restrict__ a,
    const float* __restrict__ b, const int* __restrict__ bins) {
  __shared__ float s_x[K_LUT];            // breakpoints
  __shared__ float s_ab[2 * (K_LUT + 1)]; // interleaved (a,b) pairs -> b64 gathers
  __shared__ int   s_bins[NBINS + 1];     // bucket -> breakpoint-range starts

  const int t = threadIdx.x; // blockDim.x == 256

  // ---- stage LUT to LDS with async copies (tracked by ASYNCcnt) ----
  async_ld_b128(lds_off(&s_x[4 * t]), (const v4f*)x + t); // 256*16B = 1024 floats
#pragma unroll
  for (int i = 0; i < (NBINS / 4) / 256; ++i) { // 4096 ints via b128
    int k = t + i * 256;
    async_ld_b128(lds_off(&s_bins[4 * k]), (const v4i*)bins + k);
  }
  for (int i = t; i <= K_LUT; i += 256) { // 1025 (a,b) pairs, interleaved
    async_ld_b32(lds_off(&s_ab[2 * i]), a + i);
    async_ld_b32(lds_off(&s_ab[2 * i + 1]), b + i);
  }
  if (t == 0) async_ld_b32(lds_off(&s_bins[NBINS]), bins + NBINS);

  asm volatile("s_wait_asynccnt 0x0" ::: "memory");
  __syncthreads();

  const float x0 = x[0];
  const float inv_w = (float)NBINS / (x[K_LUT - 1] - x0);

  const int stride = gridDim.x * 256;
  for (int i = blockIdx.x * 256 + t; i < nvec4; i += stride) {
    v4f d = __builtin_nontemporal_load(data4 + i);

    // Phase 1: buckets (pure VALU, 4-wide ILP)
    int j0 = bucket_of(d.x, x0, inv_w);
    int j1 = bucket_of(d.y, x0, inv_w);
    int j2 = bucket_of(d.z, x0, inv_w);
    int j3 = bucket_of(d.w, x0, inv_w);
    // Phase 2: range loads (adjacent dwords -> ds_load_2addr_b32 mergeable)
    int c0 = s_bins[j0], h0 = s_bins[j0 + 1];
    int c1 = s_bins[j1], h1 = s_bins[j1 + 1];
    int c2 = s_bins[j2], h2 = s_bins[j2 + 1];
    int c3 = s_bins[j3], h3 = s_bins[j3 + 1];
    // Phase 3: exact refine; bins avg ~0.25 breakpoints -> ~0 iterations expected
    while (c0 < h0 && s_x[c0] < d.x) ++c0;
    while (c1 < h1 && s_x[c1] < d.y) ++c1;
    while (c2 < h2 && s_x[c2] < d.z) ++c2;
    while (c3 < h3 && s_x[c3] < d.w) ++c3;
    // Phase 4: coefficient gather (8B-aligned pair -> ds_load_b64) + FMA
    v4f o;
    o.x = fmaf(s_ab[2 * c0], d.x, s_ab[2 * c0 + 1]);
    o.y = fmaf(s_ab[2 * c1], d.y, s_ab[2 * c1 + 1]);
    o.z = fmaf(s_ab[2 * c2], d.z, s_ab[2 * c2 + 1]);
    o.w = fmaf(s_ab[2 * c3], d.w, s_ab[2 * c3 + 1]);

    __builtin_nontemporal_store(o, out4 + i);
  }
}

extern "C" void kernel_launch(void* const* d_in, const int* in_sizes, int n_in,
                              void* d_out, int out_size, void* d_ws, size_t ws_size,
                              hipStream_t stream) {
  const float* data = (const float*)d_in[0]; // [4096, 16384]
  const float* x    = (const float*)d_in[1]; // [1024] sorted
  const float* a    = (const float*)d_in[2]; // [1025]
  const float* b    = (const float*)d_in[3]; // [1025]
  float* out = (float*)d_out;
  int*   bins = (int*)d_ws; // (NBINS+1) ints = 16388 bytes of scratch

  int n = in_sizes[0];
  int nvec4 = n / 4; // n = 2^26, divisible by 4

  build_bins_kernel<<<(NBINS + 1 + 255) / 256, 256, 0, stream>>>(x, bins);

  // 2048 blocks: enough waves to hide latency on every WGP; keeps redundant
  // LUT staging (~29KB * 2048 = 59MB of L2-resident reads) negligible vs the
  // 512MB HBM stream. Grid-stride covers all 16.7M float4s (~32 per thread).
  lut_eval_kernel<<<2048, 256, 0, stream>>>((const v4f*)data, (v4f*)out, nvec4,
                                            x, a, b, bins);
}